// WordAttention_58428735095102
// MI455X (gfx1250) — compile-verified
//
#include <hip/hip_runtime.h>
#include <hip/hip_bf16.h>

// Sizes from the reference
#define B_  16
#define L_  4096
#define D_  1024

typedef __attribute__((ext_vector_type(2))) float v2f;
typedef __attribute__((ext_vector_type(8))) float v8f;

// ---------------------------------------------------------------------------
// Kernel 1: dec_feature[16,1024] = dec_hidden[16,1024] @ W_feat[1024,1024]^T + b_feat
// One wave (32 lanes) per 16x16 output tile, V_WMMA_F32_16X16X4_F32, K-loop of 256.
// Layouts per CDNA5 ISA:
//   A 16x4 (f32, 2 VGPRs): lane&15 = M, VGPR0 = K+{0|2}, VGPR1 = K+{1|3} (hi half-wave -> +2)
//   B 4x16 (f32, 2 VGPRs): lane&15 = N, same K striping
//   C/D 16x16 (8 VGPRs):   VGPR r: lanes0-15 -> (M=r, N=lane), lanes16-31 -> (M=r+8, N=lane-16)
// ---------------------------------------------------------------------------
__global__ void __launch_bounds__(32)
dec_feature_wmma(const float* __restrict__ dec_hidden,
                 const float* __restrict__ W_feat,
                 const float* __restrict__ b_feat,
                 float* __restrict__ dec_feature)
{
    const int n0   = blockIdx.x * 16;         // output-column tile base
    const int lane = threadIdx.x;             // 0..31, EXEC all-1s (no divergence)
    const int m    = lane & 15;               // A row / B col within tile
    const int koff = (lane >> 4) * 2;         // 0 for lanes 0-15, 2 for lanes 16-31

    const float* __restrict__ arow = dec_hidden + m * D_;          // A row m
    const float* __restrict__ brow = W_feat + (n0 + m) * D_;       // B[k][n] = W_feat[n][k]

    v8f acc = {};
    for (int k0 = 0; k0 < D_; k0 += 4) {
        v2f a = *(const v2f*)(arow + k0 + koff);   // {A[m][k0+koff], A[m][k0+koff+1]}
        v2f b = *(const v2f*)(brow + k0 + koff);   // {B[k0+koff][n], B[k0+koff+1][n]}
        acc = __builtin_amdgcn_wmma_f32_16x16x4_f32(
            /*neg_a=*/false, a, /*neg_b=*/false, b,
            /*c_mod=*/(short)0, acc, /*reuse_a=*/false, /*reuse_b=*/false);
    }

    const int row0 = (lane < 16) ? 0 : 8;
    const int col  = n0 + (lane & 15);
    const float bias = b_feat[col];
#pragma unroll
    for (int r = 0; r < 8; ++r)
        dec_feature[(row0 + r) * D_ + col] = acc[r] + bias;
}

// ---------------------------------------------------------------------------
// Wave (32-lane) reductions
// ---------------------------------------------------------------------------
__device__ __forceinline__ float waveSum(float v) {
#pragma unroll
    for (int o = 16; o > 0; o >>= 1) v += __shfl_xor(v, o, 32);
    return v;
}
__device__ __forceinline__ float waveMax(float v) {
#pragma unroll
    for (int o = 16; o > 0; o >>= 1) v = fmaxf(v, __shfl_xor(v, o, 32));
    return v;
}

// ---------------------------------------------------------------------------
// Kernel 2: score[b,l] = sum_d tanh(enc_feature[b,l,d] + dec_feature[b,d]
//                                   + coverage[b,l]*w_cov[d]) * w_score[d]
// One wave per (b,l) row; float4 streaming over D=1024 (8 iterations/lane).
// This pass is the 256 MB enc_feature stream -> keep it single-pass, coalesced.
// ---------------------------------------------------------------------------
__global__ void __launch_bounds__(256)
score_kernel(const float* __restrict__ enc_feature,
             const float* __restrict__ dec_feature,
             const float* __restrict__ coverage,
             const float* __restrict__ w_cov,
             const float* __restrict__ w_score,
             float* __restrict__ score)
{
    const int wave = threadIdx.x >> 5;
    const int lane = threadIdx.x & 31;
    const long long row = (long long)blockIdx.x * 8 + wave;   // b*L + l
    const int b = (int)(row >> 12);                           // /4096

    const float cov = coverage[row];
    const float* __restrict__ ef = enc_feature + row * D_;
    const float* __restrict__ df = dec_feature + (long long)b * D_;

    float acc = 0.0f;
    for (int d = lane * 4; d < D_; d += 128) {
        __builtin_prefetch(ef + d + 128, 0, 0);   // global_prefetch_b8
        float4 e  = *(const float4*)(ef + d);
        float4 f  = *(const float4*)(df + d);
        float4 wc = *(const float4*)(w_cov + d);
        float4 ws = *(const float4*)(w_score + d);
        acc += tanhf(e.x + f.x + cov * wc.x) * ws.x;
        acc += tanhf(e.y + f.y + cov * wc.y) * ws.y;
        acc += tanhf(e.z + f.z + cov * wc.z) * ws.z;
        acc += tanhf(e.w + f.w + cov * wc.w) * ws.w;
    }
    acc = waveSum(acc);
    if (lane == 0) score[row] = acc;
}

// ---------------------------------------------------------------------------
// Kernel 3: per-batch masked softmax + renormalize + new_coverage.
// One block per b; L=4096 -> 16 values per thread held in registers.
// attn = exp(s-max)*mask / sum(exp(s-max)*mask); new_cov = attn + coverage.
// ---------------------------------------------------------------------------
__global__ void __launch_bounds__(256)
softmax_kernel(const float* __restrict__ score,
               const float* __restrict__ enc_mask,
               const float* __restrict__ coverage,
               float* __restrict__ attn,
               float* __restrict__ new_cov)
{
    __shared__ float red[8];
    const int b = blockIdx.x;
    const int t = threadIdx.x;
    const int wave = t >> 5, lane = t & 31;
    const long long base = (long long)b * L_;

    float vals[16];
    float lmax = -INFINITY;
#pragma unroll
    for (int i = 0; i < 16; ++i) {
        vals[i] = score[base + i * 256 + t];
        lmax = fmaxf(lmax, vals[i]);
    }
    // block max
    lmax = waveMax(lmax);
    if (lane == 0) red[wave] = lmax;
    __syncthreads();
    float bmax = red[0];
#pragma unroll
    for (int w = 1; w < 8; ++w) bmax = fmaxf(bmax, red[w]);
    __syncthreads();

    // exp * mask, accumulate masked sum
    float lsum = 0.0f;
#pragma unroll
    for (int i = 0; i < 16; ++i) {
        float m = enc_mask[base + i * 256 + t];
        vals[i] = __expf(vals[i] - bmax) * m;
        lsum += vals[i];
    }
    lsum = waveSum(lsum);
    if (lane == 0) red[wave] = lsum;
    __syncthreads();
    float total = 0.0f;
#pragma unroll
    for (int w = 0; w < 8; ++w) total += red[w];
    const float inv = 1.0f / total;

#pragma unroll
    for (int i = 0; i < 16; ++i) {
        const long long idx = base + i * 256 + t;
        const float a = vals[i] * inv;
        attn[idx]    = a;
        new_cov[idx] = a + coverage[idx];
    }
}

// ---------------------------------------------------------------------------
// Kernel 4: context[b,d] = sum_l attn[b,l] * enc_output[b,l,d]
// The second 256 MB stream. Grid (lt=16, dt=4, b=16); each block: 256 threads,
// thread owns cols dt*256+t, walks a 256-row L chunk. attn[] is wave-uniform
// -> scalar loads; enc_output reads are fully coalesced rows. Partial sums
// combined with global f32 atomics (context pre-zeroed).
// ---------------------------------------------------------------------------
__global__ void __launch_bounds__(256)
context_kernel(const float* __restrict__ enc_output,
               const float* __restrict__ attn,
               float* __restrict__ context)
{
    const int lt = blockIdx.x;   // 0..15 (L chunk)
    const int dt = blockIdx.y;   // 0..3  (D chunk)
    const int b  = blockIdx.z;   // 0..15
    const int d  = dt * 256 + threadIdx.x;

    const float* __restrict__ eo = enc_output + ((long long)b * L_ + lt * 256) * D_ + d;
    const float* __restrict__ at = attn + (long long)b * L_ + lt * 256;

    float acc = 0.0f;
#pragma unroll 4
    for (int i = 0; i < 256; ++i)
        acc += at[i] * eo[(long long)i * D_];

    atomicAdd(&context[b * D_ + d], acc);
}

// ---------------------------------------------------------------------------
// Launch
// ---------------------------------------------------------------------------
extern "C" void kernel_launch(void* const* d_in, const int* in_sizes, int n_in,
                              void* d_out, int out_size, void* d_ws, size_t ws_size,
                              hipStream_t stream) {
    const float* dec_hidden  = (const float*)d_in[0];   // [16,1024]
    const float* enc_output  = (const float*)d_in[1];   // [16,4096,1024]
    const float* enc_feature = (const float*)d_in[2];   // [16,4096,1024]
    const float* enc_mask    = (const float*)d_in[3];   // [16,4096]
    // d_in[4] = sec_attn : dead code in reference, unused
    const float* coverage    = (const float*)d_in[5];   // [16,4096]
    const float* W_feat      = (const float*)d_in[6];   // [1024,1024]
    const float* b_feat      = (const float*)d_in[7];   // [1024]
    const float* w_score     = (const float*)d_in[8];   // [1024]
    const float* w_cov       = (const float*)d_in[9];   // [1024]

    float* attn    = (float*)d_out;            // [16,4096]  = 65536
    float* context = attn + B_ * L_;           // [16,1024]  = 16384
    float* new_cov = context + B_ * D_;        // [16,4096]  = 65536

    float* dec_feature = (float*)d_ws;         // 16384 floats
    float* score       = dec_feature + B_ * D_;// 65536 floats

    // zero the atomic accumulation target (graph-capture safe)
    hipMemsetAsync(context, 0, (size_t)B_ * D_ * sizeof(float), stream);

    dec_feature_wmma<<<D_ / 16, 32, 0, stream>>>(dec_hidden, W_feat, b_feat, dec_feature);

    score_kernel<<<(B_ * L_) / 8, 256, 0, stream>>>(enc_feature, dec_feature,
                                                    coverage, w_cov, w_score, score);

    softmax_kernel<<<B_, 256, 0, stream>>>(score, enc_mask, coverage, attn, new_cov);

    context_kernel<<<dim3(16, 4, 16), 256, 0, stream>>>(enc_output, attn, context);
}